// Embedding_82532091560154
// MI455X (gfx1250) — compile-verified
//
#include <hip/hip_runtime.h>

// Embedding gather: out[row, :] = W[x[row], :]
//   x: (B*S) int32 row indices        (in_sizes[0] = 16384)
//   W: (VOCAB, D_MODEL) float32       (in_sizes[1] = 50257*1024)
//   out: (B*S, D_MODEL) float32       (out_size = 16384*1024)
//
// Pure bandwidth problem: ~128 MiB total traffic -> ~5.5 us at 23.3 TB/s.
// WMMA (one-hot matmul) would be ~1.7e15 FLOPs and stream all of W per tile:
// ~30x+ slower. Correct MI455X play: B128 vector copies, RT loads (let the
// 192MB L2 catch repeated vocab rows), NT stores (write-once output must not
// evict W from L2).

typedef float v4f __attribute__((ext_vector_type(4)));

#define D_MODEL        1024
#define F4_PER_ROW     (D_MODEL / 4)   // 256 float4 per row
#define ROWS_PER_BLOCK 4
#define THREADS        256             // 8 wave32; tid>>6 is wave-uniform

__global__ __launch_bounds__(THREADS) void embed_gather_kernel(
    const int* __restrict__ x,
    const float* __restrict__ W,
    float* __restrict__ out,
    int n_rows)
{
    const int tid  = threadIdx.x;
    const int rloc = tid >> 6;                 // 0..3, uniform within a wave32
    const int lane = tid & 63;                 // 0..63
    const int row  = blockIdx.x * ROWS_PER_BLOCK + rloc;
    if (row >= n_rows) return;

    // Wave-uniform index -> scalar load of x[row].
    const long long src = (long long)x[row] * D_MODEL;
    const long long dst = (long long)row    * D_MODEL;

    const v4f* __restrict__ wv = (const v4f*)(W + src);
    v4f*       __restrict__ ov = (v4f*)(out + dst);

    // 256 float4 per row / 64 lanes = 4 B128 transfers per lane.
    // Issue all 4 loads first: 4 in-flight global_load_b128 per lane (MLP),
    // each access fully coalesced (lane-consecutive 16B).
    v4f a = wv[lane + 0 * 64];
    v4f b = wv[lane + 1 * 64];
    v4f c = wv[lane + 2 * 64];
    v4f d = wv[lane + 3 * 64];

    // Non-temporal stores: output is never re-read; keep L2 for W rows.
    __builtin_nontemporal_store(a, ov + lane + 0 * 64);
    __builtin_nontemporal_store(b, ov + lane + 1 * 64);
    __builtin_nontemporal_store(c, ov + lane + 2 * 64);
    __builtin_nontemporal_store(d, ov + lane + 3 * 64);
}

extern "C" void kernel_launch(void* const* d_in, const int* in_sizes, int n_in,
                              void* d_out, int out_size, void* d_ws, size_t ws_size,
                              hipStream_t stream)
{
    const int*   x = (const int*)d_in[0];    // (B*S) int32 indices
    const float* W = (const float*)d_in[1];  // (VOCAB, D_MODEL) f32
    float*     out = (float*)d_out;          // (B*S, D_MODEL) f32

    const int n_rows = in_sizes[0];          // 16384
    dim3 grid((n_rows + ROWS_PER_BLOCK - 1) / ROWS_PER_BLOCK);  // 4096
    dim3 block(THREADS);

    embed_gather_kernel<<<grid, block, 0, stream>>>(x, W, out, n_rows);
}